// HierarchyGNN_32134945308693
// MI455X (gfx1250) — compile-verified
//
#include <hip/hip_runtime.h>
#include <hip/hip_bf16.h>

typedef float v2f __attribute__((ext_vector_type(2)));
typedef float v8f __attribute__((ext_vector_type(8)));

// ---------------------------------------------------------------------------
// GCN normalization helpers
// ---------------------------------------------------------------------------
__global__ void deg_init_kernel(float* __restrict__ deg, int n) {
  int i = blockIdx.x * blockDim.x + threadIdx.x;
  if (i < n) deg[i] = 1.0f;  // self-loop weight
}

__global__ void deg_accum_kernel(const long long* __restrict__ dst,
                                 const float* __restrict__ w,
                                 float* __restrict__ deg, int e) {
  int i = blockIdx.x * blockDim.x + threadIdx.x;
  if (i < e) unsafeAtomicAdd(&deg[dst[i]], w[i]);
}

__global__ void rsqrt_kernel(float* __restrict__ deg, int n) {
  int i = blockIdx.x * blockDim.x + threadIdx.x;
  if (i < n) {
    float d = deg[i];
    deg[i] = d > 0.0f ? rsqrtf(d) : 0.0f;
  }
}

// agg[n,f] = dinv[n]^2 * h[n,f]   (self-loop message, also initializes buffer)
__global__ void self_msg_kernel(const float* __restrict__ h,
                                const float* __restrict__ dinv,
                                float* __restrict__ agg, int n) {
  int i = blockIdx.x * blockDim.x + threadIdx.x;
  if (i < n * 64) {
    float di = dinv[i >> 6];
    agg[i] = di * di * h[i];
  }
}

// one wave per edge; 64 features = 2 floats per lane; fp32 HW atomics
__global__ void edge_scatter_kernel(const long long* __restrict__ srcIdx,
                                    const long long* __restrict__ dstIdx,
                                    const float* __restrict__ w,
                                    const float* __restrict__ dinv,
                                    const float* __restrict__ h,
                                    float* __restrict__ agg, int e) {
  int edge = blockIdx.x * (blockDim.x >> 5) + (threadIdx.x >> 5);
  if (edge >= e) return;
  int lane = threadIdx.x & 31;
  long long s = srcIdx[edge];
  long long d = dstIdx[edge];
  float coef = dinv[s] * w[edge] * dinv[d];
  const float* hs = h + s * 64;
  float* ad = agg + d * 64;
  unsafeAtomicAdd(&ad[lane], coef * hs[lane]);
  unsafeAtomicAdd(&ad[lane + 32], coef * hs[lane + 32]);
}

__global__ void bias_relu_kernel(float* __restrict__ agg,
                                 const float* __restrict__ bias, int n) {
  int i = blockIdx.x * blockDim.x + threadIdx.x;
  if (i < n * 64) {
    float v = agg[i] + bias[i & 63];
    agg[i] = v > 0.0f ? v : 0.0f;
  }
}

// ---------------------------------------------------------------------------
// f32 WMMA GEMM: C[M,NC] = A[M,K] * B[K,NC] (+bias)(+relu)
// Register-blocked: each wave owns one 16-row tile and computes ALL NC/16
// column tiles (one A fragment feeds NC/16 WMMAs -> 4x less A traffic).
// Each wave walks ROWT row tiles so the LDS-staged B matrix (K*NC*4 bytes
// dynamic shared) is amortized; the next tile's A rows are prefetched
// (global_prefetch_b8) while the current tile's WMMA chain executes.
// A-fragment layout (ISA 32-bit A 16x4): lane = M row; half 0 -> K,K+1;
// half 1 -> K+2,K+3.  C/D: VGPR r -> M = r (lanes 0-15) / r+8 (lanes 16-31).
// ---------------------------------------------------------------------------
template <int K, int NC, int ROWT, bool BIAS, bool RELU>
__global__ void gemm_wmma_kernel(const float* __restrict__ A,
                                 const float* __restrict__ B,
                                 const float* __restrict__ bias,
                                 float* __restrict__ C, int M) {
  constexpr int CT = NC / 16;    // column tiles per wave
  extern __shared__ float sB[];  // K * NC
  for (int i = threadIdx.x; i < K * NC; i += blockDim.x) sB[i] = B[i];
  __syncthreads();

  const int waves = blockDim.x >> 5;
  const int wave = threadIdx.x >> 5;
  const int lane = threadIdx.x & 31;
  const int half = lane >> 4;
  const int l16 = lane & 15;

  // prefetch this wave's first row tile
  {
    const int tile0 = blockIdx.x * waves * ROWT + wave;
    if (tile0 * 16 < M) {
      int pr = tile0 * 16 + l16;
      if (pr >= M) pr = M - 1;
      const float* pRow = A + (size_t)pr * K;
#pragma unroll
      for (int p = 0; p < K; p += 64) __builtin_prefetch(pRow + p, 0, 3);
    }
  }

  for (int t = 0; t < ROWT; ++t) {
    const int tile = blockIdx.x * waves * ROWT + t * waves + wave;
    const int row0 = tile * 16;
    if (row0 >= M) break;  // wave-uniform
    const int ar = (row0 + l16 < M) ? (row0 + l16) : (M - 1);
    const float* Arow = A + (size_t)ar * K;

    // prefetch the NEXT row tile's A rows behind this tile's WMMA chain
    if (t + 1 < ROWT) {
      const int ntile = blockIdx.x * waves * ROWT + (t + 1) * waves + wave;
      if (ntile * 16 < M) {
        int nr = ntile * 16 + l16;
        if (nr >= M) nr = M - 1;
        const float* nRow = A + (size_t)nr * K;
#pragma unroll
        for (int p = 0; p < K; p += 64) __builtin_prefetch(nRow + p, 0, 3);
      }
    }

    v8f acc[CT];
#pragma unroll
    for (int c = 0; c < CT; ++c) acc[c] = (v8f){};

#pragma unroll 4
    for (int k = 0; k < K; k += 4) {
      v2f a;
      a.x = Arow[k + 2 * half];
      a.y = Arow[k + 2 * half + 1];
      const float* sb0 = &sB[(k + 2 * half) * NC + l16];
      const float* sb1 = &sB[(k + 2 * half + 1) * NC + l16];
#pragma unroll
      for (int c = 0; c < CT; ++c) {
        v2f b;
        b.x = sb0[c * 16];
        b.y = sb1[c * 16];
        acc[c] = __builtin_amdgcn_wmma_f32_16x16x4_f32(
            false, a, false, b, (short)0, acc[c], false, false);
      }
    }

#pragma unroll
    for (int c = 0; c < CT; ++c) {
#pragma unroll
      for (int r = 0; r < 8; ++r) {
        int orow = row0 + r + 8 * half;
        if (orow < M) {
          float v = acc[c][r];
          if (BIAS) v += bias[c * 16 + l16];
          if (RELU) v = v > 0.0f ? v : 0.0f;
          C[(size_t)orow * NC + c * 16 + l16] = v;
        }
      }
    }
  }
}

// ---------------------------------------------------------------------------
// Fused MLP2 + softmax: logits[M,16] = H[M,64]*W[64,16] + b, softmax rows.
// One wave computes a 16x16 (nodes x classes) WMMA tile; each row's 16 class
// values live in one 16-lane half of one VGPR -> in-register softmax via
// shfl_xor masks 1,2,4,8 (stay within the half).
// ---------------------------------------------------------------------------
__global__ void mlp2_softmax_kernel(const float* __restrict__ Hid,
                                    const float* __restrict__ W,
                                    const float* __restrict__ bias,
                                    float* __restrict__ Out, int M) {
  __shared__ float sW[64 * 16];
  for (int i = threadIdx.x; i < 64 * 16; i += blockDim.x) sW[i] = W[i];
  __syncthreads();

  const int wave = threadIdx.x >> 5;
  const int lane = threadIdx.x & 31;
  const int half = lane >> 4;
  const int l16 = lane & 15;
  const int row0 = (blockIdx.x * (blockDim.x >> 5) + wave) * 16;
  if (row0 >= M) return;  // wave-uniform

  const int ar = (row0 + l16 < M) ? (row0 + l16) : (M - 1);
  const float* Hrow = Hid + (size_t)ar * 64;

  v8f acc = {};
#pragma unroll
  for (int k = 0; k < 64; k += 4) {
    v2f a, b;
    a.x = Hrow[k + 2 * half];
    a.y = Hrow[k + 2 * half + 1];
    b.x = sW[(k + 2 * half) * 16 + l16];
    b.y = sW[(k + 2 * half + 1) * 16 + l16];
    acc = __builtin_amdgcn_wmma_f32_16x16x4_f32(false, a, false, b, (short)0,
                                                acc, false, false);
  }

  const float bcls = bias[l16];
#pragma unroll
  for (int r = 0; r < 8; ++r) {
    float v = acc[r] + bcls;
    float m = v;
#pragma unroll
    for (int off = 8; off >= 1; off >>= 1) m = fmaxf(m, __shfl_xor(m, off, 32));
    float e = __expf(v - m);
    float ssum = e;
#pragma unroll
    for (int off = 8; off >= 1; off >>= 1) ssum += __shfl_xor(ssum, off, 32);
    int orow = row0 + r + 8 * half;
    if (orow < M) Out[(size_t)orow * 16 + l16] = e / ssum;
  }
}

// ---------------------------------------------------------------------------
extern "C" void kernel_launch(void* const* d_in, const int* in_sizes, int n_in,
                              void* d_out, int out_size, void* d_ws,
                              size_t ws_size, hipStream_t stream) {
  const float* x = (const float*)d_in[0];
  const long long* eidx = (const long long*)d_in[1];  // int64 [2,E]
  const float* ew = (const float*)d_in[2];
  const float* W1 = (const float*)d_in[3];
  const float* b1 = (const float*)d_in[4];
  const float* W2 = (const float*)d_in[5];
  const float* b2 = (const float*)d_in[6];
  const float* Wm1 = (const float*)d_in[7];
  const float* bm1 = (const float*)d_in[8];
  const float* Wm2 = (const float*)d_in[9];
  const float* bm2 = (const float*)d_in[10];
  float* out = (float*)d_out;

  const int N = in_sizes[0] / 256;
  const int E = in_sizes[2];
  const long long* srcIdx = eidx;      // edge_index[0]
  const long long* dstIdx = eidx + E;  // edge_index[1]

  float* ws = (float*)d_ws;
  float* dinv = ws;                     // N
  float* bufA = ws + N;                 // N*64
  float* bufB = bufA + (size_t)N * 64;  // N*64
  float* bufC = bufB + (size_t)N * 64;  // N*64

  const int TPB = 256;
  const int nBlk = (N + TPB - 1) / TPB;
  const int eBlk = (E + TPB - 1) / TPB;
  const int nfBlk = (int)(((size_t)N * 64 + TPB - 1) / TPB);
  const int rowTiles = (N + 15) / 16;
  const int ROWT = 4;                  // row tiles per wave
  const int tilesPerBlock = 4 * ROWT;  // 4 waves/block
  const int gemmBlk = (rowTiles + tilesPerBlock - 1) / tilesPerBlock;
  const int edgeWBlk = (E + 7) / 8;    // 8 waves (edges) / block

  // --- symmetric normalization (self-loops weight 1) ---
  deg_init_kernel<<<nBlk, TPB, 0, stream>>>(dinv, N);
  deg_accum_kernel<<<eBlk, TPB, 0, stream>>>(dstIdx, ew, dinv, E);
  rsqrt_kernel<<<nBlk, TPB, 0, stream>>>(dinv, N);

  // --- GCN layer 1: h = relu(agg(x @ W1) + b1) ---
  gemm_wmma_kernel<256, 64, 4, false, false>
      <<<gemmBlk, 128, 256 * 64 * 4, stream>>>(x, W1, nullptr, bufA, N);
  self_msg_kernel<<<nfBlk, TPB, 0, stream>>>(bufA, dinv, bufB, N);
  edge_scatter_kernel<<<edgeWBlk, 256, 0, stream>>>(srcIdx, dstIdx, ew, dinv,
                                                    bufA, bufB, E);
  bias_relu_kernel<<<nfBlk, TPB, 0, stream>>>(bufB, b1, N);

  // --- GCN layer 2: h = relu(agg(h @ W2) + b2) ---
  gemm_wmma_kernel<64, 64, 4, false, false>
      <<<gemmBlk, 128, 64 * 64 * 4, stream>>>(bufB, W2, nullptr, bufA, N);
  self_msg_kernel<<<nfBlk, TPB, 0, stream>>>(bufA, dinv, bufC, N);
  edge_scatter_kernel<<<edgeWBlk, 256, 0, stream>>>(srcIdx, dstIdx, ew, dinv,
                                                    bufA, bufC, E);
  bias_relu_kernel<<<nfBlk, TPB, 0, stream>>>(bufC, b2, N);

  // --- MLP hidden: relu(h @ Wm1 + bm1) ---
  gemm_wmma_kernel<64, 64, 4, true, true>
      <<<gemmBlk, 128, 64 * 64 * 4, stream>>>(bufC, Wm1, bm1, bufA, N);

  // --- logits + softmax ---
  mlp2_softmax_kernel<<<(rowTiles + 3) / 4, 128, 0, stream>>>(bufA, Wm2, bm2,
                                                              out, N);
}